// AttentionModuleBiModal_36644660969672
// MI455X (gfx1250) — compile-verified
//
#include <hip/hip_runtime.h>

// ---------------------------------------------------------------------------
// AttentionModuleBiModal, B=4, N1=N2=8192.
//
// Key identity: the 8192x8192 tanh matrix is only consumed by two weighted
// row/col reductions, so we stream it in registers. Cost model (MI455X):
// ~2*B*N*N = 5.4e8 tanh (trans pipe, native v_tanh_f32) + ~4 VALU/elem,
// with WMMA_F32_16X16X4_F32 absorbing the K-reduction adds on the matrix
// pipe (B = ones => layout-robust row sums). Memory traffic is negligible
// (inputs fit in L0/L1), so we are trans/VALU bound; all pipes co-execute.
// ---------------------------------------------------------------------------

typedef __attribute__((ext_vector_type(2))) float v2f;
typedef __attribute__((ext_vector_type(8))) float v8f;

__device__ __forceinline__ float fast_tanh(float x) {
#if __has_builtin(__builtin_amdgcn_tanhf)
  return __builtin_amdgcn_tanhf(x);           // v_tanh_f32 (gfx1250 trans op)
#elif __has_builtin(__builtin_amdgcn_tanh_f32)
  return __builtin_amdgcn_tanh_f32(x);
#else
  // tanh(x) = 1 - 2/(e^{2x}+1);  e^{2x} = 2^{x * 2*log2(e)}
  float e = __builtin_amdgcn_exp2f(x * 2.8853900817779268f);
  return 1.0f - 2.0f * __builtin_amdgcn_rcpf(e + 1.0f);
#endif
}

__device__ __forceinline__ float fast_exp(float t) {
#if __has_builtin(__builtin_amdgcn_exp2f)
  return __builtin_amdgcn_exp2f(t * 1.4426950408889634f);
#else
  return __expf(t);
#endif
}

// Generic pass:  S[r] = sum_j (ww*xc[j]+wb) * tanh( (aw*xa[r]+ab) * (cw*xc[j]+cb) )
// Each wave32 owns 16 rows (one 16-row WMMA tile); K loop covers all 8192 cols.
// tanh values (pre-scaled by the weight) are built directly in the documented
// f32 16x4 A-matrix layout; B = all-ones so D rows are exact K-sums regardless
// of B's (vgpr,half)->K mapping. f32 accumulate preserves reference precision.
__global__ __launch_bounds__(128) void pass_kernel(
    const float* __restrict__ xa, const float* __restrict__ xc,
    const float* paw, const float* pab,
    const float* pcw, const float* pcb,
    const float* pww, const float* pwb,
    float* __restrict__ S, int N) {
  const int wave = blockIdx.x * (blockDim.x >> 5) + (threadIdx.x >> 5);
  const int lane = threadIdx.x & 31;
  const int m  = lane & 15;   // A-matrix row within tile
  const int hi = lane >> 4;   // lane-half -> K pair {2*hi, 2*hi+1}
  const int r0 = wave * 16;   // first global row of this tile (N%16==0 => same batch)
  const int b  = r0 / N;

  const float aw = paw ? *paw : 1.0f;
  const float ab = pab ? *pab : 0.0f;
  const float cw = pcw ? *pcw : 1.0f;
  const float cb = pcb ? *pcb : 0.0f;
  const float ww = *pww;
  const float wb = *pwb;

  const float a = fmaf(aw, xa[r0 + m], ab);
  const float* __restrict__ col = xc + b * N;

  v8f acc0 = {0.f,0.f,0.f,0.f,0.f,0.f,0.f,0.f};
  v8f acc1 = {0.f,0.f,0.f,0.f,0.f,0.f,0.f,0.f};
  const v2f ones = {1.0f, 1.0f};

#pragma unroll 4
  for (int j0 = 0; j0 < N; j0 += 8) {
    // each lane grabs its own K pair for the two 16x4 slices
    float2 xA = *(const float2*)(col + j0 + 2 * hi);
    float2 xB = *(const float2*)(col + j0 + 4 + 2 * hi);
    v2f A0, A1;
    A0.x = fmaf(ww, xA.x, wb) * fast_tanh(a * fmaf(cw, xA.x, cb));
    A0.y = fmaf(ww, xA.y, wb) * fast_tanh(a * fmaf(cw, xA.y, cb));
    A1.x = fmaf(ww, xB.x, wb) * fast_tanh(a * fmaf(cw, xB.x, cb));
    A1.y = fmaf(ww, xB.y, wb) * fast_tanh(a * fmaf(cw, xB.y, cb));
    // D[m,n] = sum_k A[m,k] + C  (B == ones); two chains hide matrix-pipe latency
    acc0 = __builtin_amdgcn_wmma_f32_16x16x4_f32(false, A0, false, ones,
                                                 (short)0, acc0, false, false);
    acc1 = __builtin_amdgcn_wmma_f32_16x16x4_f32(false, A1, false, ones,
                                                 (short)0, acc1, false, false);
  }

  v8f r = acc0 + acc1;
  // C/D layout: VGPR v, lanes 0-15 -> M=v ; lanes 16-31 -> M=v+8 ; all N columns equal.
  if (m == 0) {
    float* o = S + r0 + hi * 8;
#pragma unroll
    for (int v = 0; v < 8; ++v) o[v] = r[v];
  }
}

// Epilogue per batch: z = sw*tanh((vw*x+vb) + S) + sb ; stable softmax over N ;
// out = softmax(z) * x
#define EPI_THREADS 512
__global__ __launch_bounds__(EPI_THREADS) void epilogue_kernel(
    const float* __restrict__ x, const float* __restrict__ S,
    const float* pvw, const float* pvb,
    const float* psw, const float* psb,
    float* __restrict__ out, int N) {
  __shared__ float zbuf[8192];
  __shared__ float red[EPI_THREADS];
  const int b = blockIdx.x;
  const float* xb = x + b * N;
  const float* Sb = S + b * N;
  float* ob = out + b * N;
  const float vw = *pvw, vb = *pvb, sw = *psw, sb = *psb;
  const int t = threadIdx.x;
  const int per = N / EPI_THREADS;  // 16

  float lmax = -3.4e38f;
#pragma unroll
  for (int k = 0; k < per; ++k) {
    int n = t + k * EPI_THREADS;
    float z = fmaf(sw, fast_tanh(fmaf(vw, xb[n], vb) + Sb[n]), sb);
    zbuf[n] = z;
    lmax = fmaxf(lmax, z);
  }
  red[t] = lmax;
  __syncthreads();
  for (int s = EPI_THREADS / 2; s > 0; s >>= 1) {
    if (t < s) red[t] = fmaxf(red[t], red[t + s]);
    __syncthreads();
  }
  const float mx = red[0];
  __syncthreads();

  float lsum = 0.0f;
#pragma unroll
  for (int k = 0; k < per; ++k) {
    int n = t + k * EPI_THREADS;
    float e = fast_exp(zbuf[n] - mx);
    zbuf[n] = e;
    lsum += e;
  }
  red[t] = lsum;
  __syncthreads();
  for (int s = EPI_THREADS / 2; s > 0; s >>= 1) {
    if (t < s) red[t] += red[t + s];
    __syncthreads();
  }
  const float inv = 1.0f / red[0];
#pragma unroll
  for (int k = 0; k < per; ++k) {
    int n = t + k * EPI_THREADS;
    ob[n] = zbuf[n] * inv * xb[n];
  }
}

extern "C" void kernel_launch(void* const* d_in, const int* in_sizes, int n_in,
                              void* d_out, int out_size, void* d_ws, size_t ws_size,
                              hipStream_t stream) {
  (void)n_in; (void)out_size; (void)ws_size;
  const float* m1 = (const float*)d_in[0];   // [B*N1]
  const float* m2 = (const float*)d_in[1];   // [B*N2]
  const float* w0 = (const float*)d_in[2];
  const float* b0 = (const float*)d_in[3];
  const float* w1 = (const float*)d_in[4];
  const float* b1 = (const float*)d_in[5];
  const float* w2 = (const float*)d_in[6];
  const float* b2 = (const float*)d_in[7];
  const float* w3 = (const float*)d_in[8];
  const float* b3 = (const float*)d_in[9];
  const float* w4 = (const float*)d_in[10];
  const float* b4 = (const float*)d_in[11];

  const int N = 8192;
  const int BN1 = in_sizes[0];  // B*N1 = 32768
  const int BN2 = in_sizes[1];  // B*N2 = 32768

  float* S1 = (float*)d_ws;     // [B*N1]
  float* S2 = S1 + BN1;         // [B*N2]
  float* out1 = (float*)d_out;  // attended_x1 [B*N1]
  float* out2 = out1 + BN1;     // attended_x2 [B*N2]

  // Pass 1: rows = i over m1.  a = m1[i];  c = w0*m2+b0;  weight = w2*m2+b2.
  int blocks1 = (BN1 / 16) / 4;  // 4 waves (16 rows each) per 128-thread block
  pass_kernel<<<blocks1, 128, 0, stream>>>(m1, m2, nullptr, nullptr, w0, b0,
                                           w2, b2, S1, N);
  // Pass 2: rows = j over m2.  a = w0*m2+b0;  c = m1[i];  weight = w1*m1+b1.
  int blocks2 = (BN2 / 16) / 4;
  pass_kernel<<<blocks2, 128, 0, stream>>>(m2, m1, w0, b0, nullptr, nullptr,
                                           w1, b1, S2, N);

  // a_m1 = tanh(w1*m1+b1 + S1); softmax(w3*a+b3); out1 = sm * m1
  epilogue_kernel<<<BN1 / N, EPI_THREADS, 0, stream>>>(m1, S1, w1, b1, w3, b3,
                                                       out1, N);
  // a_m2 = tanh(w2*m2+b2 + S2); softmax(w4*a+b4); out2 = sm * m2
  epilogue_kernel<<<BN2 / N, EPI_THREADS, 0, stream>>>(m2, S2, w2, b2, w4, b4,
                                                       out2, N);
}